// TwoWordPSDProbe_6648609374960
// MI455X (gfx1250) — compile-verified
//
#include <hip/hip_runtime.h>

// ---------------------------------------------------------------------------
// Problem constants (from reference): B=8, S=2048, D=1024, R=128
// ---------------------------------------------------------------------------
#define BB   8
#define SS   2048
#define DD   1024
#define RR   128
#define NROW (BB * SS)      // 16384 rows of the flattened (B*S, D) matrix
#define KC   64             // K-slab staged in LDS per iteration (16 k-steps)

typedef __attribute__((ext_vector_type(2))) float f32x2;
typedef __attribute__((ext_vector_type(4))) float f32x4;
typedef __attribute__((ext_vector_type(8))) float f32x8;

// D = A(16x4,f32) * B(4x16,f32) + C(16x16,f32)
// 8-arg form: (neg_a, A, neg_b, B, c_mod, C, reuse_a, reuse_b)
static __device__ __forceinline__ f32x8 wmma_f32_k4(f32x2 a, f32x2 b, f32x8 c) {
    return __builtin_amdgcn_wmma_f32_16x16x4_f32(false, a, false, b,
                                                 (short)0, c, false, false);
}

// ---------------------------------------------------------------------------
// Kernel 1: T(16384,128) = X(16384,1024) @ P(1024,128), fp32 WMMA.
//
// Workgroup = 8 waves = 128 rows. The B fragments (slices of P) are identical
// for every wave, so the workgroup cooperatively repacks a K-slab of P into
// LDS in *fragment order* once, and each wave then fetches each 4x16 B-frag
// with a single conflict-free ds_load_b64 (32 lanes x 8B = one 256B bank
// sweep). A-frags stream from global as contiguous 8B loads per lane.
//
// Fragment layouts (CDNA5 ISA 7.12.2, 32-bit, wave32):
//   A 16x4 : lanes 0-15 hold row M=lane, VGPR0=K0 VGPR1=K1;
//            lanes 16-31 hold row M=lane-16, VGPR0=K2 VGPR1=K3.
//   B 4x16 : lanes 0-15 hold col N=lane, VGPR0=K0-row VGPR1=K1-row;
//            lanes 16-31 hold col N=lane-16, VGPR0=K2 VGPR1=K3.
//   C 16x16: VGPR i -> M = i + 8*(lane>=16), N = lane&15.
// ---------------------------------------------------------------------------
__global__ __launch_bounds__(256) void proj_kernel(const float* __restrict__ X,
                                                   const float* __restrict__ P,
                                                   float* __restrict__ T) {
    // LDS: [KC/4 k-steps][8 col-tiles][32 lanes][2 floats] = 8192 f32 = 32 KB
    __shared__ float ldsP[(KC / 4) * 8 * 32 * 2];

    const int tid  = threadIdx.x;
    const int lane = tid & 31;
    const int half = lane >> 4;      // 0: K={0,1}, 1: K={2,3}
    const int lr   = lane & 15;
    const int row0 = (blockIdx.x * 8 + (tid >> 5)) * 16;

    const float* xrow = X + (size_t)(row0 + lr) * DD;

    f32x8 acc[8] = {};               // 8 col-tiles x 8 VGPRs = rows 0..15, cols 0..127

    for (int kc = 0; kc < DD; kc += KC) {
        __syncthreads();             // previous slab fully consumed
        // Cooperative repack: 4096 fragment-units, 16 per thread.
        // unit u -> (ks = u>>8, c = (u>>5)&7, lane' = u&31)
#pragma unroll
        for (int r = 0; r < 16; ++r) {
            const int u  = r * 256 + tid;
            const int ks = u >> 8;
            const int c  = (u >> 5) & 7;
            const int ul = u & 31;
            const int k  = kc + ks * 4 + 2 * (ul >> 4);
            const int nn = c * 16 + (ul & 15);
            ldsP[u * 2 + 0] = P[(size_t)k * RR + nn];        // B[K][N]
            ldsP[u * 2 + 1] = P[(size_t)(k + 1) * RR + nn];  // B[K+1][N]
        }
        __syncthreads();             // slab visible to all waves

#pragma unroll 4
        for (int ks = 0; ks < KC / 4; ++ks) {
            f32x2 a = *(const f32x2*)(xrow + kc + ks * 4 + 2 * half);
            const f32x2* bf = (const f32x2*)&ldsP[(ks * 8) * 64 + lane * 2];
#pragma unroll
            for (int c = 0; c < 8; ++c)
                acc[c] = wmma_f32_k4(a, bf[c * 32], acc[c]);
        }
    }

    // Store T tile: each half-wave writes contiguous 64B segments.
#pragma unroll
    for (int c = 0; c < 8; ++c)
#pragma unroll
        for (int i = 0; i < 8; ++i)
            T[(size_t)(row0 + i + 8 * half) * RR + c * 16 + lr] = acc[c][i];
}

// ---------------------------------------------------------------------------
// Kernel 2: sq[n] = sum_r T[n][r]^2   (one thread per row, float4 loads)
// ---------------------------------------------------------------------------
__global__ __launch_bounds__(256) void sq_kernel(const float* __restrict__ T,
                                                 float* __restrict__ sq) {
    const int n = blockIdx.x * blockDim.x + threadIdx.x;   // 0..16383
    const f32x4* row = (const f32x4*)(T + (size_t)n * RR);
    float s = 0.0f;
#pragma unroll
    for (int i = 0; i < RR / 4; ++i) {
        f32x4 v = row[i];
        s += v.x * v.x + v.y * v.y + v.z * v.z + v.w * v.w;
    }
    sq[n] = s;
}

// ---------------------------------------------------------------------------
// Kernel 3: per batch b: d2[i][j] = sq[i] + sq[j] - 2 * (Tb Tb^T)[i][j]
// One wave per 32x64 output tile: 8 WMMA accumulators fed by 2 A-frags +
// 4 B-frags per k-step (8 wmma : 6 b64 loads). For a Gram matrix the B-frag
// (= Tb^T slice) has the exact same per-lane load pattern as the A-frag,
// just a different row base, so one loader serves both.
// ---------------------------------------------------------------------------
__global__ __launch_bounds__(256) void pair_kernel(const float* __restrict__ T,
                                                   const float* __restrict__ sq,
                                                   float* __restrict__ out) {
    const int wid  = blockIdx.x * 8 + (threadIdx.x >> 5);  // 0..16383
    const int lane = threadIdx.x & 31;
    const int half = lane >> 4;
    const int lr   = lane & 15;

    const int b   = wid >> 11;        // / (64*32)
    const int rem = wid & 2047;
    const int r0  = (rem >> 5) * 32;  // row-tile base within batch  (64 tiles)
    const int c0  = (rem & 31) * 64;  // col-tile base within batch  (32 tiles)

    const float* Tb  = T  + (size_t)b * SS * RR;
    const float* sqb = sq + b * SS;
    float*       ob  = out + (size_t)b * SS * SS;

    const float* ra0 = Tb + (size_t)(r0 + lr)      * RR;
    const float* ra1 = Tb + (size_t)(r0 + 16 + lr) * RR;
    const float* rb[4] = {
        Tb + (size_t)(c0 + 0  + lr) * RR,
        Tb + (size_t)(c0 + 16 + lr) * RR,
        Tb + (size_t)(c0 + 32 + lr) * RR,
        Tb + (size_t)(c0 + 48 + lr) * RR,
    };

    f32x8 acc0[4] = {};               // rows r0..r0+15  x 4 col-tiles
    f32x8 acc1[4] = {};               // rows r0+16..+31 x 4 col-tiles

#pragma unroll 4
    for (int k = 0; k < RR; k += 4) {
        const int off = k + 2 * half;
        f32x2 a0 = *(const f32x2*)(ra0 + off);
        f32x2 a1 = *(const f32x2*)(ra1 + off);
#pragma unroll
        for (int ci = 0; ci < 4; ++ci) {
            f32x2 bf = *(const f32x2*)(rb[ci] + off);
            acc0[ci] = wmma_f32_k4(a0, bf, acc0[ci]);
            acc1[ci] = wmma_f32_k4(a1, bf, acc1[ci]);
        }
    }

    float sn[4];
#pragma unroll
    for (int ci = 0; ci < 4; ++ci) sn[ci] = sqb[c0 + ci * 16 + lr];

#pragma unroll
    for (int i = 0; i < 8; ++i) {
        const int m0 = r0 + i + 8 * half;  // row of acc element i (row-tile 0)
        const int m1 = m0 + 16;            // row-tile 1
        const float sm0 = sqb[m0];
        const float sm1 = sqb[m1];
#pragma unroll
        for (int ci = 0; ci < 4; ++ci) {
            ob[(size_t)m0 * SS + c0 + ci * 16 + lr] = sm0 + sn[ci] - 2.0f * acc0[ci][i];
            ob[(size_t)m1 * SS + c0 + ci * 16 + lr] = sm1 + sn[ci] - 2.0f * acc1[ci][i];
        }
    }
}

// ---------------------------------------------------------------------------
// Launch: proj -> sq -> pair, all on `stream` (same stream => ordered).
// Workspace layout: T (16384*128 f32 = 8 MB) | sq (16384 f32 = 64 KB).
// ---------------------------------------------------------------------------
extern "C" void kernel_launch(void* const* d_in, const int* in_sizes, int n_in,
                              void* d_out, int out_size, void* d_ws, size_t ws_size,
                              hipStream_t stream) {
    const float* X = (const float*)d_in[0];   // batch (8,2048,1024) f32
    const float* P = (const float*)d_in[1];   // proj  (1024,128)   f32
    float* out = (float*)d_out;               // (8,2048,2048) f32
    float* T   = (float*)d_ws;                // (16384,128)
    float* sqv = T + (size_t)NROW * RR;       // (16384)

    // 128 blocks x 8 waves, one wave per 16-row strip
    proj_kernel<<<NROW / 128, 256, 0, stream>>>(X, P, T);
    // 16384 threads
    sq_kernel<<<NROW / 256, 256, 0, stream>>>(T, sqv);
    // 8 * 64 * 32 = 16384 waves = 2048 blocks of 8 waves
    pair_kernel<<<(BB * (SS / 32) * (SS / 64)) / 8, 256, 0, stream>>>(T, sqv, out);
}